// yololoss_44246753084155
// MI455X (gfx1250) — compile-verified
//
#include <hip/hip_runtime.h>
#include <hip/hip_bf16.h>
#include <math.h>

typedef __attribute__((ext_vector_type(2))) float v2f;
typedef __attribute__((ext_vector_type(8))) float v8f;

#define NTOT   8192
#define GG     49
#define CH     30
#define CELLS  (NTOT * GG)       // 401408
#define BLOCK  256
#define NBLK   (CELLS / BLOCK)   // 1568 (exact)
#define NPAD   1600              // next multiple of 64 -> 25 branch-free WMMA chunks
#define SLAB   (BLOCK * CH)      // 7680 floats per tensor per block

// ---------------------------------------------------------------------------
// Kernel 1: stage each block's pred/target slab into LDS with CDNA5 async
// global->LDS b128 copies (coalesced, no VGPR round-trip), then one thread
// per cell computes the loss from LDS; deterministic block reduction.
// ---------------------------------------------------------------------------
__global__ __launch_bounds__(BLOCK) void yolo_loss_partial(
    const float* __restrict__ pred, const float* __restrict__ target,
    float* __restrict__ partial)
{
    __shared__ float sh[2 * SLAB];
    __shared__ float sdata[BLOCK / 32];
    float* shp = sh;
    float* sht = sh + SLAB;

    const int tid = threadIdx.x;

    // Zero the finalize kernel's pad slots (block 0 only; own slots, no race).
    if (blockIdx.x == 0 && tid < (NPAD - NBLK))
        partial[NBLK + tid] = 0.0f;

    const long base = (long)blockIdx.x * SLAB;   // element offset of this slab
    const float* gp = pred   + base;
    const float* gt = target + base;

    // Async stage: 16B per lane per op, fully coalesced. ASYNCcnt-tracked.
    for (int off = tid * 4; off < SLAB; off += BLOCK * 4) {
        unsigned lp = (unsigned)(unsigned long long)(size_t)(&shp[off]);
        unsigned lt = (unsigned)(unsigned long long)(size_t)(&sht[off]);
        unsigned long long ap = (unsigned long long)(size_t)(&gp[off]);
        unsigned long long at = (unsigned long long)(size_t)(&gt[off]);
        asm volatile("global_load_async_to_lds_b128 %0, %1, off"
                     :: "v"(lp), "v"(ap) : "memory");
        asm volatile("global_load_async_to_lds_b128 %0, %1, off"
                     :: "v"(lt), "v"(at) : "memory");
    }
    asm volatile("s_wait_asynccnt 0x0" ::: "memory");  // my wave's copies done
    __syncthreads();                                    // everyone's visible

    const float* p = shp + tid * CH;
    const float* t = sht + tid * CH;

    float tp[CH], tt[CH];
    #pragma unroll
    for (int i = 0; i < CH; ++i) { tp[i] = p[i]; tt[i] = t[i]; }

    const float conf = tt[4];
    const float cof  = (conf > 0.0f)  ? 1.0f : 0.0f;
    const float nof  = (conf == 0.0f) ? 1.0f : 0.0f;

    // loss_no = 0.5 * nof * ((p4-t4)^2 + (p9-t9)^2)
    float d4 = tp[4] - tt[4];
    float d9 = tp[9] - tt[9];
    float loss = 0.5f * nof * (d4 * d4 + d9 * d9);

    // loss_class = cof * sum_{c=10..29} (p-t)^2
    float lc = 0.0f;
    #pragma unroll
    for (int c = 10; c < CH; ++c) { float d = tp[c] - tt[c]; lc += d * d; }
    loss += cof * lc;

    // pair IoU of each pred box vs target box 0, replicating the reference's
    // py1 == py0 bug (y-extent 0 => area_p = area_t = 100).
    const float tx0 = tt[0] - tt[2] * 0.5f;
    const float tx1 = tt[0] + tt[2] * 0.5f;
    const float ty0 = tt[1] - tt[3] * 0.5f;   // ty1 == ty0

    float iou[2];
    #pragma unroll
    for (int b = 0; b < 2; ++b) {
        const float* pb = tp + 5 * b;
        float px0 = pb[0] - pb[2] * 0.5f;
        float px1 = pb[0] + pb[2] * 0.5f;
        float py0 = pb[1] - pb[3] * 0.5f;     // py1 == py0
        float ix  = fminf(px1, tx1) - fmaxf(px0, tx0);
        float iy  = fminf(py0, ty0) - fmaxf(py0, ty0); // = -|py0 - ty0|
        float inter = ix * iy;
        iou[b] = inter / (200.0f - inter);
    }
    const int r = (iou[1] > iou[0]) ? 1 : 0;   // argmax, tie -> box 0
    const int o = 1 - r;

    const float* pr = tp + 5 * r;
    const float* tr = tt + 5 * r;
    const float* po = tp + 5 * o;
    const float* to = tt + 5 * o;

    // loss_xy
    float dx = tr[0] - pr[0];
    float dy = tr[1] - pr[1];
    loss += 5.0f * cof * (dx * dx + dy * dy);

    // loss_wh (sqrt guard matches where(co, ., 1))
    if (cof > 0.0f) {
        float sw = sqrtf(tr[2]) - sqrtf(pr[2]);
        float sh2 = sqrtf(tr[3]) - sqrtf(pr[3]);
        loss += 5.0f * (sw * sw + sh2 * sh2);
    }

    // loss_c + loss_nore
    float dc = tr[4] - pr[4];
    float dn = to[4] - po[4];
    loss += cof * (dc * dc + 0.5f * dn * dn);

    // Deterministic reduction: shfl tree within wave32, LDS across 8 waves.
    float v = loss;
    #pragma unroll
    for (int off = 16; off > 0; off >>= 1) v += __shfl_down(v, off, 32);

    if ((tid & 31) == 0) sdata[tid >> 5] = v;
    __syncthreads();
    if (tid == 0) {
        float s = 0.0f;
        #pragma unroll
        for (int i = 0; i < BLOCK / 32; ++i) s += sdata[i];
        partial[blockIdx.x] = s;
    }
}

// ---------------------------------------------------------------------------
// Kernel 2: single wave32, branch-free. WMMA sum of 1600 padded partials:
// D = ones(16x4) x B(4x16) + C folds 64 values per issue into 16 replicated
// column accumulators (layout-independent for a pure sum). EXEC all-1s.
// ---------------------------------------------------------------------------
__global__ __launch_bounds__(32) void yolo_loss_finalize(
    const float* __restrict__ partial, float* __restrict__ out)
{
    const int lane = threadIdx.x;

    v2f a; a.x = 1.0f; a.y = 1.0f;   // A = ones (16x4 f32)
    v8f acc = {};                     // C/D accumulator

    #pragma unroll 1
    for (int c = 0; c < NPAD / 64; ++c) {
        const int base = c * 64;
        v2f b;
        b.x = partial[base + lane];
        b.y = partial[base + 32 + lane];
        // 8 args: (neg_a, A, neg_b, B, c_mod, C, reuse_a, reuse_b)
        acc = __builtin_amdgcn_wmma_f32_16x16x4_f32(
            false, a, false, b, (short)0, acc, false, false);
    }

    // Row 0 of D: acc[0] lanes 0..15 (lanes 16..31 hold duplicated row M=8).
    float v = (lane < 16) ? acc[0] : 0.0f;
    #pragma unroll
    for (int off = 16; off > 0; off >>= 1) v += __shfl_down(v, off, 32);

    if (lane == 0) out[0] = v / (float)NTOT;
}

// ---------------------------------------------------------------------------
extern "C" void kernel_launch(void* const* d_in, const int* in_sizes, int n_in,
                              void* d_out, int out_size, void* d_ws, size_t ws_size,
                              hipStream_t stream)
{
    (void)in_sizes; (void)n_in; (void)out_size; (void)ws_size;
    const float* pred   = (const float*)d_in[0];
    const float* target = (const float*)d_in[1];
    float* out     = (float*)d_out;
    float* partial = (float*)d_ws;   // NPAD floats = 6400 bytes

    yolo_loss_partial<<<NBLK, BLOCK, 0, stream>>>(pred, target, partial);
    yolo_loss_finalize<<<1, 32, 0, stream>>>(partial, out);
}